// BaselineDNN_50697793962160
// MI455X (gfx1250) — compile-verified
//
#include <hip/hip_runtime.h>

#define B_ 2048
#define L_ 200
#define V_ 50000
#define D_ 300
#define H_ 128
#define C_ 20

typedef float v2f __attribute__((ext_vector_type(2)));
typedef float v8f __attribute__((ext_vector_type(8)));

// ---------------------------------------------------------------------------
// Kernel 1: embedding gather + sum/max reduction over L, produce rep[B, 2D].
// rep[b, 0:300]   = sum_l emb[x[b,l], :] / lengths[b]^2   (reference's bug kept)
// rep[b, 300:600] = max_l emb[x[b,l], :]                  (over full padded L)
// One block per batch row; 256 threads map to d (stride-1 => coalesced B32).
// ---------------------------------------------------------------------------
__global__ __launch_bounds__(256) void gather_reduce_kernel(
    const int* __restrict__ x, const int* __restrict__ lengths,
    const float* __restrict__ emb, float* __restrict__ rep)
{
    const int b   = blockIdx.x;
    const int tid = threadIdx.x;

    __shared__ int idx[L_];
    for (int i = tid; i < L_; i += 256) idx[i] = x[(size_t)b * L_ + i];
    __syncthreads();

    const int d0 = tid;         // always < 300
    const int d1 = tid + 256;   // valid for tid < 44
    float s0 = 0.f, s1 = 0.f;
    float m0 = -__builtin_inff(), m1 = -__builtin_inff();

    for (int l = 0; l < L_; ++l) {
        const float* row = emb + (size_t)idx[l] * D_;
        if (l + 1 < L_) {
            // hide random-row latency: gfx1250 global_prefetch_b8 of next row
            const float* nrow = emb + (size_t)idx[l + 1] * D_;
            __builtin_prefetch(nrow + d0, 0, 1);
        }
        float v0 = row[d0];
        s0 += v0; m0 = fmaxf(m0, v0);
        if (d1 < D_) {
            float v1 = row[d1];
            s1 += v1; m1 = fmaxf(m1, v1);
        }
    }

    const float len = (float)lengths[b];
    const float inv = 1.f / (len * len);
    float* out = rep + (size_t)b * (2 * D_);
    out[d0]      = s0 * inv;
    out[D_ + d0] = m0;
    if (d1 < D_) {
        out[d1]      = s1 * inv;
        out[D_ + d1] = m1;
    }
}

// ---------------------------------------------------------------------------
// Kernel 2: fused GEMM1 (+bias,ReLU) and GEMM2 (+bias) via V_WMMA_F32_16X16X4_F32.
// One wave (32 threads) per 16-row batch tile.
//   GEMM1: h[16,128]  = relu(rep_tile[16,600] @ W_new^T + b_new)  -> LDS
//   GEMM2: out[16,20] = h[16,128] @ W3^T + b3   (N padded to 32, masked store)
// ---------------------------------------------------------------------------
__global__ __launch_bounds__(32) void gemm_wmma_kernel(
    const float* __restrict__ rep, const float* __restrict__ Wn,
    const float* __restrict__ bn,  const float* __restrict__ W3,
    const float* __restrict__ b3,  float* __restrict__ out)
{
    const int lane  = threadIdx.x;        // 0..31 (wave32)
    const int b0    = blockIdx.x * 16;    // batch tile base row
    const int mrow  = lane & 15;          // M (or N for B-matrix) within tile
    const int kb    = (lane >> 4) * 2;    // K base for this half-wave: 0 or 2
    const int rbase = (lane >> 4) * 8;    // C/D row base for this half-wave

    __shared__ float hbuf[16 * H_];       // h tile, row-major [16][128]

    // ---- GEMM1: K = 600, N = 128 (8 n-tiles), A reused across n-tiles ----
    v8f acc[H_ / 16];
    #pragma unroll
    for (int nt = 0; nt < H_ / 16; ++nt)
        #pragma unroll
        for (int v = 0; v < 8; ++v) acc[nt][v] = 0.f;

    const float* arow = rep + (size_t)(b0 + mrow) * (2 * D_);
    for (int k0 = 0; k0 < 2 * D_; k0 += 4) {
        v2f a;
        a.x = arow[k0 + kb];
        a.y = arow[k0 + kb + 1];
        #pragma unroll
        for (int nt = 0; nt < H_ / 16; ++nt) {
            const int n = nt * 16 + mrow;
            v2f bb;                       // B[k][n] = W_new[n][k]
            bb.x = Wn[(size_t)n * (2 * D_) + k0 + kb];
            bb.y = Wn[(size_t)n * (2 * D_) + k0 + kb + 1];
            acc[nt] = __builtin_amdgcn_wmma_f32_16x16x4_f32(
                false, a, false, bb, (short)0, acc[nt], false, false);
        }
    }

    // bias + ReLU, stage h tile in LDS (row-major) for the A-side of GEMM2
    #pragma unroll
    for (int nt = 0; nt < H_ / 16; ++nt) {
        const int col  = nt * 16 + mrow;
        const float bv = bn[col];
        #pragma unroll
        for (int v = 0; v < 8; ++v) {
            float val = acc[nt][v] + bv;
            hbuf[(rbase + v) * H_ + col] = fmaxf(val, 0.f);
        }
    }
    __syncthreads();

    // ---- GEMM2: K = 128, N = 20 padded to 32 (2 n-tiles) ----
    v8f acc2[2];
    #pragma unroll
    for (int nt = 0; nt < 2; ++nt)
        #pragma unroll
        for (int v = 0; v < 8; ++v) acc2[nt][v] = 0.f;

    for (int k0 = 0; k0 < H_; k0 += 4) {
        v2f a;
        a.x = hbuf[mrow * H_ + k0 + kb];
        a.y = hbuf[mrow * H_ + k0 + kb + 1];
        #pragma unroll
        for (int nt = 0; nt < 2; ++nt) {
            const int n = nt * 16 + mrow;
            const bool valid = (n < C_);   // zero-pad columns 20..31
            v2f bb;
            bb.x = valid ? W3[(size_t)n * H_ + k0 + kb]     : 0.f;
            bb.y = valid ? W3[(size_t)n * H_ + k0 + kb + 1] : 0.f;
            acc2[nt] = __builtin_amdgcn_wmma_f32_16x16x4_f32(
                false, a, false, bb, (short)0, acc2[nt], false, false);
        }
    }

    // bias + masked store of logits[16, 20]
    #pragma unroll
    for (int nt = 0; nt < 2; ++nt) {
        const int col = nt * 16 + mrow;
        if (col < C_) {
            const float bv = b3[col];
            #pragma unroll
            for (int v = 0; v < 8; ++v)
                out[(size_t)(b0 + rbase + v) * C_ + col] = acc2[nt][v] + bv;
        }
    }
}

extern "C" void kernel_launch(void* const* d_in, const int* in_sizes, int n_in,
                              void* d_out, int out_size, void* d_ws, size_t ws_size,
                              hipStream_t stream) {
    const int*   x       = (const int*)d_in[0];    // [B, L]
    const int*   lengths = (const int*)d_in[1];    // [B]
    const float* emb     = (const float*)d_in[2];  // [V, D]
    const float* Wn      = (const float*)d_in[3];  // [H, 2D]
    const float* bn      = (const float*)d_in[4];  // [H]
    const float* W3      = (const float*)d_in[5];  // [C, H]
    const float* b3      = (const float*)d_in[6];  // [C]
    float*       outp    = (float*)d_out;          // [B, C]

    float* rep = (float*)d_ws;                     // [B, 2D] = 4.9 MB scratch

    gather_reduce_kernel<<<B_, 256, 0, stream>>>(x, lengths, emb, rep);
    gemm_wmma_kernel<<<B_ / 16, 32, 0, stream>>>(rep, Wn, bn, W3, b3, outp);
}